// OrbitalCofactorAntiequivarianceLayer_20031727468619
// MI455X (gfx1250) — compile-verified
//
#include <hip/hip_runtime.h>
#include <math.h>

typedef float v2f __attribute__((ext_vector_type(2)));
typedef float v8f __attribute__((ext_vector_type(8)));

#define WAVES_PER_BLOCK 8

// One wave handles one (b,s) 16x16 matrix:
//   1) y = xs(16x256) @ W_s(256x16) + b_s  via 64x V_WMMA_F32_16X16X4_F32
//      (two independent accumulators to break the WMMA->WMMA RAW chain)
//   2) env(n,o) = sum_i exp(-||env_dim[s,i,o] @ r(n,i)||) * env_ion[s,i,o]
//   3) M = y * env  -> LDS (stride-17 rows, bank-conflict free)
//   4) cof_n = M[n,0] * det(M) * invM[0,n], via one lane-parallel pivoted LU
__global__ __launch_bounds__(32 * WAVES_PER_BLOCK)
void orbital_cof_kernel(const float* __restrict__ eq,
                        const float* __restrict__ rei,
                        const float* __restrict__ W,
                        const float* __restrict__ bvec,
                        const float* __restrict__ edim,
                        const float* __restrict__ eion,
                        float* __restrict__ out,
                        int nmat)
{
    __shared__ float sM[WAVES_PER_BLOCK][16 * 17];

    const int tid  = threadIdx.x;
    const int wave = tid >> 5;
    const int lane = tid & 31;
    const int idx  = lane & 15;   // A-row / B-col / C-col (o)
    const int half = lane >> 4;

    const int m = blockIdx.x * WAVES_PER_BLOCK + wave;   // m = b*2 + s
    if (m >= nmat) return;
    const int s = m & 1;

    // ---------------- GEMM: 16x256 @ 256x16 via WMMA f32 16x16x4 ----------------
    const float* Abase = eq + (size_t)m * (16 * 256);        // xs[b,s,:,:]
    const float* Wbase = W  + (size_t)s * (256 * 16);        // W[s,:,:]
    const float* arow  = Abase + idx * 256 + 2 * half;       // row idx, K pair at 2*half
    const float* wcol  = Wbase + 2 * half * 16 + idx;        // col idx, K pair at 2*half

    v8f c0 = {0.f, 0.f, 0.f, 0.f, 0.f, 0.f, 0.f, 0.f};
    v8f c1 = {0.f, 0.f, 0.f, 0.f, 0.f, 0.f, 0.f, 0.f};

    #pragma unroll 2
    for (int kk = 0; kk < 32; ++kk) {
        // even K-step -> c0
        v2f a0 = *(const v2f*)(arow + (2 * kk) * 4);
        v2f b0;
        b0.x = wcol[(2 * kk) * 64];
        b0.y = wcol[(2 * kk) * 64 + 16];
        c0 = __builtin_amdgcn_wmma_f32_16x16x4_f32(false, a0, false, b0,
                                                   (short)0, c0, false, false);
        // odd K-step -> c1 (independent of c0 chain)
        v2f a1 = *(const v2f*)(arow + (2 * kk + 1) * 4);
        v2f b1;
        b1.x = wcol[(2 * kk + 1) * 64];
        b1.y = wcol[(2 * kk + 1) * 64 + 16];
        c1 = __builtin_amdgcn_wmma_f32_16x16x4_f32(false, a1, false, b1,
                                                   (short)0, c1, false, false);
    }

    const float bias = bvec[s * 16 + idx];

    // ---------------- envelope: env[n=v+8*half][o=idx] ----------------
    float env[8];
    #pragma unroll
    for (int v = 0; v < 8; ++v) env[v] = 0.f;

    const float* reibase  = rei  + (size_t)m * 16 * (8 * 3);   // [n][i][3]
    const float* edimbase = edim + (size_t)s * 8 * (16 * 9);   // [i][o][3][3]
    const float* eionbase = eion + (size_t)s * 8 * 16;         // [i][o]

    #pragma unroll
    for (int i = 0; i < 8; ++i) {
        const float* E = edimbase + (i * 16 + idx) * 9;
        const float e00 = E[0], e01 = E[1], e02 = E[2];
        const float e10 = E[3], e11 = E[4], e12 = E[5];
        const float e20 = E[6], e21 = E[7], e22 = E[8];
        const float ion = eionbase[i * 16 + idx];
        #pragma unroll
        for (int v = 0; v < 8; ++v) {
            const int n = v + 8 * half;
            const float* rp = reibase + (n * 8 + i) * 3;
            const float r0 = rp[0], r1 = rp[1], r2 = rp[2];
            const float a0 = e00 * r0 + e01 * r1 + e02 * r2;
            const float a1 = e10 * r0 + e11 * r1 + e12 * r2;
            const float a2 = e20 * r0 + e21 * r1 + e22 * r2;
            const float nrm = sqrtf(a0 * a0 + a1 * a1 + a2 * a2);
            env[v] += expf(-nrm) * ion;
        }
    }

    // ---------------- M = (y + b) * env -> LDS (row stride 17) ----------------
    float* slot = sM[wave];
    #pragma unroll
    for (int v = 0; v < 8; ++v) {
        const float mval = (c0[v] + c1[v] + bias) * env[v];
        slot[(v + 8 * half) * 17 + idx] = mval;
    }

    __syncthreads();

    // ---------------- lane-parallel LU of M^T with partial pivoting ----------
    // lane r (both halves duplicate) owns row r of A = M^T: A[r][c] = M[c][r]
    const int r = idx;
    float a[16];
    #pragma unroll
    for (int cc = 0; cc < 16; ++cc) a[cc] = slot[cc * 17 + r];
    float rhs  = (r == 0) ? 1.f : 0.f;   // solving M^T x = e0
    float det  = 1.f;
    float sign = 1.f;

    #pragma unroll
    for (int k = 0; k < 16; ++k) {
        // argmax |a[k]| over rows r >= k (ties -> smallest index)
        float cand = (r >= k) ? fabsf(a[k]) : -1.f;
        int   pidx = r;
        #pragma unroll
        for (int off = 8; off >= 1; off >>= 1) {
            const float ov = __shfl_xor(cand, off, 16);
            const int   oi = __shfl_xor(pidx, off, 16);
            const bool take = (ov > cand) || (ov == cand && oi < pidx);
            cand = take ? ov : cand;
            pidx = take ? oi : pidx;
        }
        sign = (pidx != k) ? -sign : sign;

        // swap lanes k <-> pidx; only entries j >= k are live afterwards
        const int src = (r == k) ? pidx : ((r == pidx) ? k : r);
        #pragma unroll
        for (int j = k; j < 16; ++j) a[j] = __shfl(a[j], src, 16);
        rhs = __shfl(rhs, src, 16);

        const float pk = __shfl(a[k], k, 16);   // pivot (uniform in group)
        det *= pk;
        const float mfac = (r > k) ? (a[k] / pk) : 0.f;
        #pragma unroll
        for (int j = k + 1; j < 16; ++j) {
            const float pj = __shfl(a[j], k, 16);
            a[j] = fmaf(-mfac, pj, a[j]);
        }
        const float prhs = __shfl(rhs, k, 16);
        rhs = fmaf(-mfac, prhs, rhs);
    }
    det *= sign;

    // back substitution: x[k] broadcast as produced; lane r keeps x[r]
    float x = 0.f;
    #pragma unroll
    for (int k = 15; k >= 0; --k) {
        const float akk = __shfl(a[k], k, 16);
        const float xk  = __shfl(rhs, k, 16) / akk;
        x   = (r == k) ? xk : x;
        rhs = (r < k) ? fmaf(-a[k], xk, rhs) : rhs;
    }

    // cof_r = M[r][0] * det(M) * invM[0][r]
    const float m_r0 = slot[r * 17 + 0];
    const float cof  = m_r0 * det * x;
    if (half == 0) out[(size_t)m * 16 + r] = cof;
}

extern "C" void kernel_launch(void* const* d_in, const int* in_sizes, int n_in,
                              void* d_out, int out_size, void* d_ws, size_t ws_size,
                              hipStream_t stream) {
    const float* eq   = (const float*)d_in[0];   // (B, 32, 256)
    const float* rei  = (const float*)d_in[1];   // (B, 32, 8, 3)
    const float* W    = (const float*)d_in[2];   // (2, 256, 16)
    const float* bvec = (const float*)d_in[3];   // (2, 16)
    const float* edim = (const float*)d_in[4];   // (2, 8, 16, 3, 3)
    const float* eion = (const float*)d_in[5];   // (2, 8, 16)
    float* out = (float*)d_out;                  // (B, 2, 16, 1)

    const int B    = in_sizes[0] / (32 * 256);
    const int nmat = B * 2;
    const int blocks = (nmat + WAVES_PER_BLOCK - 1) / WAVES_PER_BLOCK;

    orbital_cof_kernel<<<blocks, 32 * WAVES_PER_BLOCK, 0, stream>>>(
        eq, rei, W, bvec, edim, eion, out, nmat);
}